// GraphSAGE_74895639707856
// MI455X (gfx1250) — compile-verified
//
#include <hip/hip_runtime.h>
#include <math.h>

typedef __attribute__((ext_vector_type(2))) float v2f;
typedef __attribute__((ext_vector_type(4))) float v4f;
typedef __attribute__((ext_vector_type(8))) float v8f;

#define N_NODES   100000
#define N_EDGES   1600000
#define D_FEAT    128      // K dim for both layers (D_IN == D_HID == 128)

// ---------------------------------------------------------------- utilities
__global__ void zero_kernel(float* __restrict__ p, int n) {
    int i = blockIdx.x * blockDim.x + threadIdx.x;
    int stride = gridDim.x * blockDim.x;
    for (; i < n; i += stride) p[i] = 0.0f;
}

__global__ void deg_kernel(const int* __restrict__ dst, float* __restrict__ deg, int E) {
    int i = blockIdx.x * blockDim.x + threadIdx.x;
    int stride = gridDim.x * blockDim.x;
    for (; i < E; i += stride) {
        __hip_atomic_fetch_add(&deg[dst[i]], 1.0f,
                               __ATOMIC_RELAXED, __HIP_MEMORY_SCOPE_AGENT);
    }
}

// One wave per edge: 32 lanes x float4 = 128 features. Gather is one fully
// coalesced 512B read; scatter is 4 f32 atomics per lane into L2-resident agg.
__global__ void scatter_kernel(const float* __restrict__ x,
                               const int* __restrict__ src,
                               const int* __restrict__ dst,
                               float* __restrict__ agg, int E) {
    int lane = threadIdx.x & 31;
    int wid  = (blockIdx.x * blockDim.x + threadIdx.x) >> 5;
    int nw   = (gridDim.x * blockDim.x) >> 5;
    for (int e = wid; e < E; e += nw) {
        int s = src[e];
        int d = dst[e];
        v4f v = *(const v4f*)(x + (size_t)s * D_FEAT + lane * 4);
        float* base = agg + (size_t)d * D_FEAT + lane * 4;
#pragma unroll
        for (int i = 0; i < 4; ++i) {
            __hip_atomic_fetch_add(base + i, v[i],
                                   __ATOMIC_RELAXED, __HIP_MEMORY_SCOPE_AGENT);
        }
    }
}

// ------------------------------------------------------------ fused SAGE GEMM
// out[m, n] = act( (agg[m,:]/max(deg[m],1)) @ wl + xin[m,:] @ wr + bias[n] )
// One wave computes one 16x16 output tile with V_WMMA_F32_16X16X4_F32,
// accumulating both matrix terms into a single f32 accumulator.
template <int DN, bool RELU>
__global__ void sage_gemm_kernel(const float* __restrict__ agg,
                                 const float* __restrict__ deg,
                                 const float* __restrict__ xin,
                                 const float* __restrict__ wl,
                                 const float* __restrict__ wr,
                                 const float* __restrict__ bias,
                                 float* __restrict__ out) {
    constexpr int NT = DN / 16;
    constexpr int TOTAL_TILES = (N_NODES / 16) * NT;

    int wid = (blockIdx.x * blockDim.x + threadIdx.x) >> 5;
    if (wid >= TOTAL_TILES) return;   // wave-uniform exit; EXEC all-ones inside

    int mtile = wid / NT;
    int ntile = wid % NT;
    int lane  = threadIdx.x & 31;
    int m     = lane & 15;            // A-matrix row held by this lane
    int half  = lane >> 4;            // 0 -> K+0/K+1, 1 -> K+2/K+3
    int row   = mtile * 16 + m;
    int col   = ntile * 16 + m;       // B-matrix column held by this lane

    float rd = 1.0f / fmaxf(deg[row], 1.0f);
    const float* arow = agg + (size_t)row * D_FEAT;
    const float* xrow = xin + (size_t)row * D_FEAT;

    v8f c = {};
    for (int k = 0; k < D_FEAT; k += 4) {
        int kk = k + half * 2;
        v2f am, ax, bl, br;
        am.x = arow[kk] * rd;       am.y = arow[kk + 1] * rd;   // fused mean
        ax.x = xrow[kk];            ax.y = xrow[kk + 1];
        bl.x = wl[(size_t)kk * DN + col];
        bl.y = wl[(size_t)(kk + 1) * DN + col];
        br.x = wr[(size_t)kk * DN + col];
        br.y = wr[(size_t)(kk + 1) * DN + col];
        c = __builtin_amdgcn_wmma_f32_16x16x4_f32(false, am, false, bl,
                                                  (short)0, c, false, false);
        c = __builtin_amdgcn_wmma_f32_16x16x4_f32(false, ax, false, br,
                                                  (short)0, c, false, false);
    }

    float bcol = bias[col];
#pragma unroll
    for (int j = 0; j < 8; ++j) {
        int r = mtile * 16 + j + half * 8;          // C layout: VGPR j -> M=j / M=j+8
        float v = c[j] + bcol;
        if (RELU) v = fmaxf(v, 0.0f);
        out[(size_t)r * DN + ntile * 16 + (lane & 15)] = v;
    }
}

// ------------------------------------------------------- row-wise log_softmax
__global__ void logsoftmax_kernel(float* __restrict__ out) {
    int r = blockIdx.x * blockDim.x + threadIdx.x;
    if (r >= N_NODES) return;
    float* p = out + (size_t)r * 16;
    float v[16];
    float mx = -3.4e38f;
#pragma unroll
    for (int i = 0; i < 16; ++i) { v[i] = p[i]; mx = fmaxf(mx, v[i]); }
    float s = 0.0f;
#pragma unroll
    for (int i = 0; i < 16; ++i) s += expf(v[i] - mx);
    float lse = mx + logf(s);
#pragma unroll
    for (int i = 0; i < 16; ++i) p[i] = v[i] - lse;
}

// ----------------------------------------------------------------- dispatcher
extern "C" void kernel_launch(void* const* d_in, const int* in_sizes, int n_in,
                              void* d_out, int out_size, void* d_ws, size_t ws_size,
                              hipStream_t stream) {
    (void)in_sizes; (void)n_in; (void)out_size; (void)ws_size;

    const float* x    = (const float*)d_in[0];
    const int*   edge = (const int*)  d_in[1];   // (2, E) row-major
    const float* w1l  = (const float*)d_in[2];
    const float* w1r  = (const float*)d_in[3];
    const float* b1   = (const float*)d_in[4];
    const float* w2l  = (const float*)d_in[5];
    const float* w2r  = (const float*)d_in[6];
    const float* b2   = (const float*)d_in[7];

    const int* src = edge;            // edge_index[0]
    const int* dst = edge + N_EDGES;  // edge_index[1]

    float* ws  = (float*)d_ws;
    float* agg = ws;                                   // N * 128
    float* deg = agg + (size_t)N_NODES * D_FEAT;       // N
    float* h   = deg + N_NODES;                        // N * 128
    float* out = (float*)d_out;                        // N * 16

    const int AGG_N = N_NODES * D_FEAT;

    // ---- layer 1 ----
    zero_kernel<<<4096, 256, 0, stream>>>(agg, AGG_N + N_NODES);   // agg + deg
    deg_kernel<<<2048, 256, 0, stream>>>(dst, deg, N_EDGES);
    scatter_kernel<<<8192, 256, 0, stream>>>(x, src, dst, agg, N_EDGES);
    // tiles = 6250 * 8 = 50000 waves, 8 waves/block
    sage_gemm_kernel<128, true><<<6250, 256, 0, stream>>>(agg, deg, x, w1l, w1r, b1, h);

    // ---- layer 2 (degree unchanged, reuse deg) ----
    zero_kernel<<<4096, 256, 0, stream>>>(agg, AGG_N);
    scatter_kernel<<<8192, 256, 0, stream>>>(h, src, dst, agg, N_EDGES);
    // tiles = 6250 waves
    sage_gemm_kernel<16, false><<<(6250 + 7) / 8, 256, 0, stream>>>(agg, deg, h, w2l, w2r, b2, out);

    logsoftmax_kernel<<<(N_NODES + 255) / 256, 256, 0, stream>>>(out);
}